// Convolution_5583457484920
// MI455X (gfx1250) — compile-verified
//
#include <hip/hip_runtime.h>
#include <hip/hip_bf16.h>
#include <math.h>

#define NN 10000
#define NE 320000

typedef __attribute__((ext_vector_type(2))) float v2f;
typedef __attribute__((ext_vector_type(8))) float v8f;

#define STR 68                 // padded row stride for 16x64 LDS tile
#define WAVES_PB 8
#define EDGES_PW 16
#define EDGES_PB (WAVES_PB * EDGES_PW)   // 128 edges per block; 320000/128 = 2500 blocks

// Branchless tanh-gelu: 0.5*x*(1+tanh(z)) == x * sigmoid(2z) == x / (1 + exp(-2z)).
// exp(-2z) = exp2(-2*log2(e)*z) -> single v_exp_f32; division -> v_rcp_f32.
__device__ __forceinline__ float gelu_fast(float x) {
    float z = 0.7978845608028654f * x * __builtin_fmaf(0.044715f, x * x, 1.0f);
    float e = __builtin_amdgcn_exp2f(-2.8853900817779268f * z);  // exp(-2z)
    return x * __builtin_amdgcn_rcpf(1.0f + e);
}

// B operand (4x16 micro-tile) of row-major [K][ncols] matrix in global memory.
// VGPR v, lane l -> B[k*4 + v + 2*(l>=16)][t*16 + (l&15)]
__device__ __forceinline__ v2f load_b(const float* __restrict__ w, int ncols,
                                      int k, int t, int lane) {
    int hi  = lane >> 4;
    int col = t * 16 + (lane & 15);
    int row = k * 4 + 2 * hi;
    v2f b;
    b.x = w[row * ncols + col];
    b.y = w[(row + 1) * ncols + col];
    return b;
}

// A operand (16x4 micro-tile) from LDS tile H[16][STR].
// VGPR v, lane l -> A[l&15][k*4 + v + 2*(l>=16)]
__device__ __forceinline__ v2f load_a_lds(const float* H, int k, int lane) {
    int hi  = lane >> 4;
    int row = lane & 15;
    int c   = k * 4 + 2 * hi;
    v2f a;
    a.x = H[row * STR + c];
    a.y = H[row * STR + c + 1];
    return a;
}

// ---------------- zero accumulators ----------------
__global__ void zero_kernel(float* __restrict__ p, int n) {
    int i = blockIdx.x * blockDim.x + threadIdx.x;
    if (i < n) p[i] = 0.0f;
}

// ---------------- node linear 1: feat/self split ----------------
__global__ __launch_bounds__(256) void node_lin1_kernel(
    const float* __restrict__ ni, const float* __restrict__ W0e,
    const float* __restrict__ W1o,
    float* __restrict__ feat0, float* __restrict__ self0,
    float* __restrict__ feat1, float* __restrict__ self1) {
    __shared__ float row[128];
    int n = blockIdx.x;
    int r = threadIdx.x;
    if (r < 128) row[r] = ni[n * 128 + r];
    __syncthreads();
    const float sc = 0.17677669529663687f;  // 1/sqrt(32)
    if (r < 64) {
        int w = r;
        float acc = 0.f;
#pragma unroll
        for (int u = 0; u < 32; ++u) acc += row[u] * W0e[u * 64 + w];
        acc *= sc;
        if (w < 32) feat0[n * 32 + w] = acc;
        else        self0[n * 32 + (w - 32)] = acc;
    } else {
        int q = r - 64;            // 0..191
        int w = q / 3, i = q - 3 * w;
        float acc = 0.f;
#pragma unroll
        for (int u = 0; u < 32; ++u) acc += row[32 + u * 3 + i] * W1o[u * 64 + w];
        acc *= sc;
        if (w < 32) feat1[n * 96 + w * 3 + i] = acc;
        else        self1[n * 96 + (w - 32) * 3 + i] = acc;
    }
}

// ---------------- edge kernel: MLP (WMMA f32) + tensor product + scatter ----------------
__global__ __launch_bounds__(256) void edge_kernel(
    const float* __restrict__ es,   // (E,8) edge_scalar_attr
    const float* __restrict__ ea,   // (E,4) edge_attr
    const int*   __restrict__ esrc, const int* __restrict__ edst,
    const float* __restrict__ Wm1,  // (8,64)
    const float* __restrict__ Wm2,  // (64,64)
    const float* __restrict__ Wp00, const float* __restrict__ Wp01,
    const float* __restrict__ Wp10, const float* __restrict__ Wp11,
    const float* __restrict__ feat0, const float* __restrict__ feat1,
    float* __restrict__ n0, float* __restrict__ n1) {
    __shared__ float smem[WAVES_PB * 16 * STR];
    int tid  = threadIdx.x;
    int wave = tid >> 5;
    int lane = tid & 31;
    int hi   = lane >> 4;
    float* H = smem + wave * 16 * STR;
    int ebase = blockIdx.x * EDGES_PB + wave * EDGES_PW;

    // ---- GEMM1: h1 = gelu( S(16x8) @ Wm1(8x64) * 1/sqrt(8) ); store h1*0.125 into H
    int erow = ebase + (lane & 15);
    if (erow > NE - 1) erow = NE - 1;          // clamp (no divergence around WMMA)
    const float inv_sqrt8 = 0.35355339059327373f;
#pragma unroll
    for (int t = 0; t < 4; ++t) {
        v8f acc = {};
#pragma unroll
        for (int k = 0; k < 2; ++k) {
            int c0 = k * 4 + 2 * hi;
            v2f a;
            a.x = es[erow * 8 + c0]     * inv_sqrt8;
            a.y = es[erow * 8 + c0 + 1] * inv_sqrt8;
            v2f b = load_b(Wm1, 64, k, t, lane);
            acc = __builtin_amdgcn_wmma_f32_16x16x4_f32(false, a, false, b,
                                                        (short)0, acc, false, false);
        }
#pragma unroll
        for (int v = 0; v < 8; ++v)
            H[(v + 8 * hi) * STR + t * 16 + (lane & 15)] =
                gelu_fast(acc[v]) * 0.125f;     // fold 1/sqrt(64) of GEMM2 here
    }
    __syncthreads();

    // ---- GEMM2: h2 = gelu( (h1/8) @ Wm2(64x64) ), kept in registers
    v8f h2[4];
#pragma unroll
    for (int t = 0; t < 4; ++t) {
        v8f acc = {};
#pragma unroll
        for (int k = 0; k < 16; ++k) {
            v2f a = load_a_lds(H, k, lane);
            v2f b = load_b(Wm2, 64, k, t, lane);
            acc = __builtin_amdgcn_wmma_f32_16x16x4_f32(false, a, false, b,
                                                        (short)0, acc, false, false);
        }
        h2[t] = acc;
    }
    __syncthreads();
#pragma unroll
    for (int t = 0; t < 4; ++t)
#pragma unroll
        for (int v = 0; v < 8; ++v)
            H[(v + 8 * hi) * STR + t * 16 + (lane & 15)] = gelu_fast(h2[t][v]);
    __syncthreads();

    // ---- GEMM3: w_p = h2(16x64) @ Wp_p(64x32), p = {00,01,10,11}
    v8f wacc[4][2];
    const float* Wp[4] = {Wp00, Wp01, Wp10, Wp11};
#pragma unroll
    for (int p = 0; p < 4; ++p)
#pragma unroll
        for (int t = 0; t < 2; ++t) {
            v8f acc = {};
#pragma unroll
            for (int k = 0; k < 16; ++k) {
                v2f a = load_a_lds(H, k, lane);
                v2f b = load_b(Wp[p], 32, k, t, lane);
                acc = __builtin_amdgcn_wmma_f32_16x16x4_f32(false, a, false, b,
                                                            (short)0, acc, false, false);
            }
            wacc[p][t] = acc;
        }

    // ---- tensor product + scatter-add (1/sqrt(2) * 1/sqrt(32) = 0.125 folded)
    const float inv_sqrt3 = 0.5773502691896258f;
#pragma unroll
    for (int t = 0; t < 2; ++t) {
        int u = t * 16 + (lane & 15);
#pragma unroll
        for (int v = 0; v < 8; ++v) {
            int e = ebase + v + 8 * hi;
            if (e < NE) {
                int s = esrc[e], d = edst[e];
                float a0 = ea[e * 4 + 0], a1 = ea[e * 4 + 1];
                float a2 = ea[e * 4 + 2], a3 = ea[e * 4 + 3];
                float e0  = feat0[s * 32 + u];
                float e1x = feat1[s * 96 + u * 3 + 0];
                float e1y = feat1[s * 96 + u * 3 + 1];
                float e1z = feat1[s * 96 + u * 3 + 2];
                float w00v = wacc[0][t][v], w01v = wacc[1][t][v];
                float w10v = wacc[2][t][v], w11v = wacc[3][t][v];

                unsafeAtomicAdd(&n0[d * 64 + u], 0.125f * w00v * e0 * a0);
                float dot = e1x * a1 + e1y * a2 + e1z * a3;
                unsafeAtomicAdd(&n0[d * 64 + 32 + u],
                                0.125f * inv_sqrt3 * w11v * dot);
                float c01 = 0.125f * w01v * e0;
                unsafeAtomicAdd(&n1[d * 192 + u * 3 + 0], c01 * a1);
                unsafeAtomicAdd(&n1[d * 192 + u * 3 + 1], c01 * a2);
                unsafeAtomicAdd(&n1[d * 192 + u * 3 + 2], c01 * a3);
                float c10 = 0.125f * w10v * a0;
                unsafeAtomicAdd(&n1[d * 192 + (32 + u) * 3 + 0], c10 * e1x);
                unsafeAtomicAdd(&n1[d * 192 + (32 + u) * 3 + 1], c10 * e1y);
                unsafeAtomicAdd(&n1[d * 192 + (32 + u) * 3 + 2], c10 * e1z);
            }
        }
    }
}

// ---------------- node linear 2 + mixing ----------------
__global__ __launch_bounds__(128) void node_out_kernel(
    const float* __restrict__ n0, const float* __restrict__ n1,
    const float* __restrict__ W20e, const float* __restrict__ W21o,
    const float* __restrict__ self0, const float* __restrict__ self1,
    float* __restrict__ out) {
    int n = blockIdx.x, r = threadIdx.x;
    const float cc = 0.9238795325112867f;   // cos(pi/8)
    const float ss = 0.3826834323650898f;   // sin(pi/8)
    const float sc = 0.125f;                // 1/sqrt(64)
    if (r < 32) {
        float acc = 0.f;
#pragma unroll
        for (int k = 0; k < 64; ++k) acc += n0[n * 64 + k] * W20e[k * 32 + r];
        out[n * 128 + r] = cc * self0[n * 32 + r] + ss * (acc * sc);
    } else {
        int q = r - 32;                      // 0..95
        int w = q / 3, i = q - 3 * w;
        float acc = 0.f;
#pragma unroll
        for (int u = 0; u < 64; ++u) acc += n1[n * 192 + u * 3 + i] * W21o[u * 32 + w];
        out[n * 128 + r] = cc * self1[n * 96 + q] + ss * (acc * sc);
    }
}

extern "C" void kernel_launch(void* const* d_in, const int* in_sizes, int n_in,
                              void* d_out, int out_size, void* d_ws, size_t ws_size,
                              hipStream_t stream) {
    const float* node_input = (const float*)d_in[0];
    const float* edge_attr  = (const float*)d_in[1];
    const float* edge_scal  = (const float*)d_in[2];
    const float* W1_0e      = (const float*)d_in[3];
    const float* W1_1o      = (const float*)d_in[4];
    const float* W_mlp1     = (const float*)d_in[5];
    const float* W_mlp2     = (const float*)d_in[6];
    const float* Wp_00      = (const float*)d_in[7];
    const float* Wp_01      = (const float*)d_in[8];
    const float* Wp_10      = (const float*)d_in[9];
    const float* Wp_11      = (const float*)d_in[10];
    const float* W2_0e      = (const float*)d_in[11];
    const float* W2_1o      = (const float*)d_in[12];
    const int*   edge_src   = (const int*)d_in[13];
    const int*   edge_dst   = (const int*)d_in[14];

    float* ws    = (float*)d_ws;
    float* feat0 = ws;                    // N*32
    float* self0 = feat0 + NN * 32;       // N*32
    float* feat1 = self0 + NN * 32;       // N*96
    float* self1 = feat1 + NN * 96;       // N*96
    float* n0    = self1 + NN * 96;       // N*64
    float* n1    = n0    + NN * 64;       // N*192

    // zero the scatter accumulators (n0 and n1 are contiguous: N*256 floats)
    int nz = NN * 256;
    zero_kernel<<<(nz + 255) / 256, 256, 0, stream>>>(n0, nz);

    node_lin1_kernel<<<NN, 256, 0, stream>>>(node_input, W1_0e, W1_1o,
                                             feat0, self0, feat1, self1);

    edge_kernel<<<NE / EDGES_PB, 256, 0, stream>>>(
        edge_scal, edge_attr, edge_src, edge_dst,
        W_mlp1, W_mlp2, Wp_00, Wp_01, Wp_10, Wp_11,
        feat0, feat1, n0, n1);

    node_out_kernel<<<NN, 128, 0, stream>>>(n0, n1, W2_0e, W2_1o,
                                            self0, self1, (float*)d_out);
}